// AttackRTransformerLayer_82248623719017
// MI455X (gfx1250) — compile-verified
//
#include <hip/hip_runtime.h>
#include <math.h>

// Problem constants (match reference)
#define Bn   8
#define Sn   128
#define HIDn 256
#define Hn   4
#define DHn  64
#define FFn  1024

// GEMM macro-tile config: 256 threads (8 waves), 64x64 output, K-block 32.
#define MT   64
#define NT   64
#define KBk  32
#define APAD 36   // A panel row stride in floats (64 rows x 32 cols, padded)
#define BPAD 68   // B panel row stride in floats (32 rows x 64 cols, padded)

typedef float v2f __attribute__((ext_vector_type(2)));
typedef float v8f __attribute__((ext_vector_type(8)));
typedef __attribute__((address_space(3))) float lds_float;

// V_WMMA_F32_16X16X4_F32: D(16x16 f32) = A(16x4 f32) x B(4x16 f32) + C
static __device__ __forceinline__ v8f wmma4(v2f a, v2f b, v8f c) {
  return __builtin_amdgcn_wmma_f32_16x16x4_f32(false, a, false, b, (short)0, c,
                                               false, false);
}

// LDS byte offset (relative to workgroup LDS base) of a __shared__ object.
static __device__ __forceinline__ unsigned int lds_off(void* p) {
  return (unsigned int)(size_t)(lds_float*)p;
}

// 16-byte async copy global -> LDS (ASYNCcnt-tracked, gfx1250).
static __device__ __forceinline__ void async_copy16(unsigned int ldsoff,
                                                    const float* g) {
  asm volatile("global_load_async_to_lds_b128 %0, %1, off"
               :: "v"(ldsoff), "v"((unsigned long long)(size_t)g)
               : "memory");
}

static __device__ __forceinline__ float gelu_exact(float x) {
  return 0.5f * x * (1.0f + erff(x * 0.70710678118654752440f));
}

// ---------------------------------------------------------------------------
// Cooperative WMMA GEMM: C[M,N] = A[M,K] @ B[K,N] (+bias) (gelu?) (+residual)
// Block = 256 threads = 8 waves; 64x64 macro-tile (2 16x16 tiles per wave).
// A/B panels double-buffered in LDS, filled by global_load_async_to_lds_b128
// and pipelined with s_wait_asynccnt. M,N multiples of 64; K multiple of 32.
// WMMA fragment layouts per CDNA5 ISA 7.12.2 (wave32):
//   A 16x4 : lanes 0-15 -> M=lane, VGPR0=K0,VGPR1=K1; lanes 16-31 -> K2,K3
//   B 4x16 : lanes 0-15 -> N=lane, K0/K1; lanes 16-31 -> K2/K3
//   C 16x16: acc[r] -> M = r + 8*(lane>=16), N = lane&15
// ---------------------------------------------------------------------------
__global__ void gemm_wmma_kernel(const float* __restrict__ A, int lda,
                                 const float* __restrict__ Bm, int ldb,
                                 const float* __restrict__ bias,
                                 const float* __restrict__ residual, int ldr,
                                 float* __restrict__ C, int ldc,
                                 int K, int apply_gelu) {
  __shared__ __align__(16) float sA[2][MT * APAD];
  __shared__ __align__(16) float sB[2][KBk * BPAD];

  const int tid  = threadIdx.x;      // 0..255
  const int wave = tid >> 5;
  const int lane = tid & 31;
  const int half = lane >> 4;
  const int l    = lane & 15;

  const int rowBase = blockIdx.y * MT;
  const int colBase = blockIdx.x * NT;

  // Two 16x16 output tiles per wave over the 4x4 tile grid.
  const int tm0 = ((wave)     >> 2) * 16, tn0 = ((wave)     & 3) * 16;
  const int tm1 = ((wave + 8) >> 2) * 16, tn1 = ((wave + 8) & 3) * 16;
  v8f acc0 = {};
  v8f acc1 = {};

  const int nkb = K / KBk;

  // Stage k-block kbi into LDS buffer buf: 4 async b128 issues per thread.
  auto issue = [&](int kbi, int buf) {
    const int kBase = kbi * KBk;
    for (int c = tid; c < (MT * KBk) / 4; c += 256) {   // 2 iters
      const int row  = c >> 3;          // KBk/4 = 8 chunks per A row
      const int col4 = (c & 7) * 4;
      async_copy16(lds_off(&sA[buf][row * APAD + col4]),
                   A + (size_t)(rowBase + row) * lda + kBase + col4);
    }
    for (int c = tid; c < (KBk * NT) / 4; c += 256) {   // 2 iters
      const int row  = c >> 4;          // NT/4 = 16 chunks per B row
      const int col4 = (c & 15) * 4;
      async_copy16(lds_off(&sB[buf][row * BPAD + col4]),
                   Bm + (size_t)(kBase + row) * ldb + colBase + col4);
    }
  };

  issue(0, 0);
  for (int i = 0; i < nkb; ++i) {
    const int buf = i & 1;
    if (i + 1 < nkb) {
      issue(i + 1, buf ^ 1);
      // 4 newest async issues (next block) may remain in flight.
      asm volatile("s_wait_asynccnt 0x4" ::: "memory");
    } else {
      asm volatile("s_wait_asynccnt 0x0" ::: "memory");
    }
    __syncthreads();  // all waves' panel-i data resident in LDS

    const float* pa = &sA[buf][0];
    const float* pb = &sB[buf][0];
#pragma unroll
    for (int kb = 0; kb < KBk; kb += 4) {
      v2f a0, a1, b0, b1;
      a0.x = pa[(tm0 + l) * APAD + kb + 2 * half];
      a0.y = pa[(tm0 + l) * APAD + kb + 2 * half + 1];
      a1.x = pa[(tm1 + l) * APAD + kb + 2 * half];
      a1.y = pa[(tm1 + l) * APAD + kb + 2 * half + 1];
      b0.x = pb[(kb + 2 * half) * BPAD + tn0 + l];
      b0.y = pb[(kb + 2 * half + 1) * BPAD + tn0 + l];
      b1.x = pb[(kb + 2 * half) * BPAD + tn1 + l];
      b1.y = pb[(kb + 2 * half + 1) * BPAD + tn1 + l];
      acc0 = wmma4(a0, b0, acc0);
      acc1 = wmma4(a1, b1, acc1);
    }
    __syncthreads();  // done reading this buffer before it is refilled
  }

  // Epilogue: bias -> (gelu) -> (+residual) -> store, for both tiles.
  const int n0 = colBase + tn0 + l;
  const int n1 = colBase + tn1 + l;
  const float bv0 = bias ? bias[n0] : 0.0f;
  const float bv1 = bias ? bias[n1] : 0.0f;
#pragma unroll
  for (int r = 0; r < 8; ++r) {
    const int m0 = rowBase + tm0 + r + 8 * half;
    const int m1 = rowBase + tm1 + r + 8 * half;
    float v0 = acc0[r] + bv0;
    float v1 = acc1[r] + bv1;
    if (apply_gelu) { v0 = gelu_exact(v0); v1 = gelu_exact(v1); }
    if (residual) {
      v0 += residual[(size_t)m0 * ldr + n0];
      v1 += residual[(size_t)m1 * ldr + n1];
    }
    C[(size_t)m0 * ldc + n0] = v0;
    C[(size_t)m1 * ldc + n1] = v1;
  }
}

// ---------------------------------------------------------------------------
// Compose bias through second affine: outb[n] = sum_k bsrc[k]*AW[k,n] + Ab[n]
// ---------------------------------------------------------------------------
__global__ void bias_compose_kernel(const float* __restrict__ bsrc,
                                    const float* __restrict__ AW,
                                    const float* __restrict__ Ab,
                                    float* __restrict__ outb) {
  const int n = threadIdx.x;
  float s = Ab[n];
  for (int kk = 0; kk < HIDn; ++kk) s += bsrc[kk] * AW[(size_t)kk * HIDn + n];
  outb[n] = s;
}

// Row softmax over S elements held in LDS (optionally scale + mask first).
static __device__ __forceinline__ void row_softmax(float* __restrict__ r,
                                                   float scale,
                                                   const float* __restrict__ mrow) {
  float mx = -1e30f;
  for (int j = 0; j < Sn; ++j) {
    float v = r[j] * scale + (mrow ? mrow[j] : 0.0f);
    r[j] = v;
    mx = fmaxf(mx, v);
  }
  float s = 0.0f;
  for (int j = 0; j < Sn; ++j) { float e = expf(r[j] - mx); r[j] = e; s += e; }
  const float is = 1.0f / s;
  for (int j = 0; j < Sn; ++j) r[j] *= is;
}

// ---------------------------------------------------------------------------
// Fused attention for one (b,h): scores + attack scores via WMMA into LDS,
// rich/order/dist terms, 5 softmax stages in LDS, ctx = final @ vh via WMMA.
// Block = 256 threads = 8 waves. Dynamic LDS = 3*S*S + 4*S floats (~198 KB,
// legal on CDNA5: 320 KB LDS per WGP/workgroup).
// ---------------------------------------------------------------------------
__global__ void attention_kernel(const float* __restrict__ q,
                                 const float* __restrict__ k,
                                 const float* __restrict__ v,
                                 const float* __restrict__ aq,
                                 const float* __restrict__ ak,
                                 const float* __restrict__ mask,
                                 const float* __restrict__ order_w,
                                 const float* __restrict__ order_b,
                                 const float* __restrict__ dist_w,
                                 const float* __restrict__ dist_b,
                                 const float* __restrict__ scalar,
                                 float* __restrict__ ctx) {
  extern __shared__ float smem[];
  float* sc = smem;                       // origin scores -> final probs
  float* rc = smem + Sn * Sn;             // rich scores  -> combined probs
  float* at = smem + 2 * Sn * Sn;         // attack scores -> attack probs
  float* qo = smem + 3 * Sn * Sn;         // [S] order proj (q side)
  float* ko = qo + Sn;                    // [S] (k side)
  float* qd = ko + Sn;                    // [S] dist proj (q side)
  float* kd = qd + Sn;                    // [S] (k side)

  const int bh = blockIdx.x;
  const int bb = bh / Hn;
  const int hh = bh % Hn;
  const size_t base = (size_t)bb * Sn * HIDn + (size_t)hh * DHn;
  const float* qB  = q  + base;
  const float* kB  = k  + base;
  const float* vB  = v  + base;
  const float* aqB = aq + base;
  const float* akB = ak + base;
  const float* mB  = mask + (size_t)bb * Sn * Sn;

  const int tid  = threadIdx.x;
  const int wave = tid >> 5;
  const int lane = tid & 31;
  const int half = lane >> 4;
  const int l    = lane & 15;

  // Per-row projections q.order_w[:DH], k.order_w[DH:], same for dist_w.
  if (tid < Sn) {
    float s1 = 0.f, s2 = 0.f, s3 = 0.f, s4 = 0.f;
    const float* qr = qB + (size_t)tid * HIDn;
    const float* kr = kB + (size_t)tid * HIDn;
    for (int d = 0; d < DHn; ++d) {
      s1 += qr[d] * order_w[d];
      s2 += kr[d] * order_w[DHn + d];
      s3 += qr[d] * dist_w[d];
      s4 += kr[d] * dist_w[DHn + d];
    }
    qo[tid] = s1; ko[tid] = s2; qd[tid] = s3; kd[tid] = s4;
  }

  // scores = qh @ kh^T and attack = aqh @ akh^T (8x8 tiles, wave w = tile row w)
  const int row0 = wave * 16;
  for (int tc = 0; tc < 8; ++tc) {
    const int col0 = tc * 16;
    v8f acc = {};
    v8f acc2 = {};
    const float* qp  = qB  + (size_t)(row0 + l) * HIDn + 2 * half;
    const float* kp  = kB  + (size_t)(col0 + l) * HIDn + 2 * half;  // B[d][j]=k[j][d]
    const float* aqp = aqB + (size_t)(row0 + l) * HIDn + 2 * half;
    const float* akp = akB + (size_t)(col0 + l) * HIDn + 2 * half;
    for (int kb = 0; kb < DHn; kb += 4) {
      v2f a, b;
      a.x = qp[kb];  a.y = qp[kb + 1];
      b.x = kp[kb];  b.y = kp[kb + 1];
      acc = wmma4(a, b, acc);
      a.x = aqp[kb]; a.y = aqp[kb + 1];
      b.x = akp[kb]; b.y = akp[kb + 1];
      acc2 = wmma4(a, b, acc2);
    }
#pragma unroll
    for (int r = 0; r < 8; ++r) {
      sc[(row0 + r + 8 * half) * Sn + col0 + l] = acc[r];
      at[(row0 + r + 8 * half) * Sn + col0 + l] = acc2[r];
    }
  }
  __syncthreads();

  // rich scores = scores + err_order + err_dist (elementwise over S*S)
  const float s2c = scalar[0] * scalar[0] * 0.5f;
  const float ob = order_b[0];
  const float db = dist_b[0];
  for (int idx = tid; idx < Sn * Sn; idx += 256) {
    const int i = idx >> 7;
    const int j = idx & (Sn - 1);
    const float pr = 1.0f / (1.0f + expf(-(qo[i] + ko[j] + ob)));
    const float go = (j > i) ? 1.0f : 0.0f;
    const float eo = logf(pr + 1e-24f) * go + logf(1.0f - pr + 1e-24f) * (1.0f - go);
    const float gd = logf(fabsf((float)(j - i)) + 1.0f);
    const float pd = qd[i] + kd[j] + db;
    rc[idx] = sc[idx] + eo - (gd - pd) * (gd - pd) * s2c;
  }
  __syncthreads();

  // Five softmax stages, one row per thread (threads 0..127).
  const float inv = 0.125f;  // 1/sqrt(DH)
  if (tid < Sn) {
    float* r_sc = sc + (size_t)tid * Sn;
    float* r_rc = rc + (size_t)tid * Sn;
    float* r_at = at + (size_t)tid * Sn;
    const float* mrow = mB + (size_t)tid * Sn;
    row_softmax(r_rc, inv, mrow);               // rich_probs
    row_softmax(r_sc, inv, mrow);               // origin_probs
    for (int j = 0; j < Sn; ++j) r_rc[j] = r_sc[j] + 0.5f * r_rc[j];
    row_softmax(r_rc, 1.0f, nullptr);           // combined
    row_softmax(r_at, inv, mrow);               // attack_probs
    for (int j = 0; j < Sn; ++j) r_sc[j] = r_rc[j] + 0.5f * r_at[j];
    row_softmax(r_sc, 1.0f, nullptr);           // final_probs
  }
  __syncthreads();

  // ctx = final_probs(SxS, LDS) @ vh(SxDH). 8x4 tiles, wave w = tile row w.
  for (int tc = 0; tc < 4; ++tc) {
    const int col0 = tc * 16;
    v8f acc = {};
    const float* apl = sc + (size_t)(row0 + l) * Sn + 2 * half;
    const float* bp  = vB + (size_t)(2 * half) * HIDn + col0 + l;
    for (int kb = 0; kb < Sn; kb += 4) {
      v2f a, b;
      a.x = apl[kb];                  a.y = apl[kb + 1];
      b.x = bp[(size_t)kb * HIDn];    b.y = bp[(size_t)(kb + 1) * HIDn];
      acc = wmma4(a, b, acc);
    }
#pragma unroll
    for (int r = 0; r < 8; ++r)
      ctx[((size_t)bb * Sn + row0 + r + 8 * half) * HIDn + hh * DHn + col0 + l] = acc[r];
  }
}

// ---------------------------------------------------------------------------
// Row LayerNorm over HID=256 columns; one block (256 threads) per row.
// ---------------------------------------------------------------------------
__global__ void layernorm_kernel(const float* __restrict__ in,
                                 const float* __restrict__ g,
                                 const float* __restrict__ be,
                                 float* __restrict__ out) {
  __shared__ float red[HIDn];
  const int row = blockIdx.x;
  const int tid = threadIdx.x;
  const float v = in[(size_t)row * HIDn + tid];
  red[tid] = v;
  __syncthreads();
  for (int s = HIDn / 2; s > 0; s >>= 1) {
    if (tid < s) red[tid] += red[tid + s];
    __syncthreads();
  }
  const float mean = red[0] * (1.0f / HIDn);
  __syncthreads();
  const float d = v - mean;
  red[tid] = d * d;
  __syncthreads();
  for (int s = HIDn / 2; s > 0; s >>= 1) {
    if (tid < s) red[tid] += red[tid + s];
    __syncthreads();
  }
  const float var = red[0] * (1.0f / HIDn);
  out[(size_t)row * HIDn + tid] = d * rsqrtf(var + 1e-12f) * g[tid] + be[tid];
}

// ---------------------------------------------------------------------------
extern "C" void kernel_launch(void* const* d_in, const int* in_sizes, int n_in,
                              void* d_out, int out_size, void* d_ws, size_t ws_size,
                              hipStream_t stream) {
  const float* x    = (const float*)d_in[0];
  const float* mask = (const float*)d_in[1];
  const float* Wq   = (const float*)d_in[2];
  const float* bq   = (const float*)d_in[3];
  const float* Wk   = (const float*)d_in[4];
  const float* bk   = (const float*)d_in[5];
  const float* Wv   = (const float*)d_in[6];
  const float* bv   = (const float*)d_in[7];
  const float* order_w = (const float*)d_in[8];
  const float* order_b = (const float*)d_in[9];
  const float* dist_w  = (const float*)d_in[10];
  const float* dist_b  = (const float*)d_in[11];
  const float* scalar  = (const float*)d_in[12];
  const float* AWq  = (const float*)d_in[13];
  const float* Abq  = (const float*)d_in[14];
  const float* AWk  = (const float*)d_in[15];
  const float* Abk  = (const float*)d_in[16];
  const float* Wd   = (const float*)d_in[17];
  const float* bd   = (const float*)d_in[18];
  const float* g1   = (const float*)d_in[19];
  const float* be1  = (const float*)d_in[20];
  const float* W1   = (const float*)d_in[21];
  const float* b1   = (const float*)d_in[22];
  const float* W2   = (const float*)d_in[23];
  const float* b2   = (const float*)d_in[24];
  const float* g2   = (const float*)d_in[25];
  const float* be2  = (const float*)d_in[26];
  float* out = (float*)d_out;

  const int M  = Bn * Sn;  // 1024 token rows
  float* ws = (float*)d_ws;
  float* q   = ws;                 // [M,HID]
  float* k   = q   + (size_t)M * HIDn;
  float* v   = k   + (size_t)M * HIDn;
  float* aq  = v   + (size_t)M * HIDn;
  float* ak  = aq  + (size_t)M * HIDn;
  float* CWq = ak  + (size_t)M * HIDn;      // [HID,HID]
  float* CWk = CWq + (size_t)HIDn * HIDn;
  float* cbq = CWk + (size_t)HIDn * HIDn;   // [HID]
  float* cbk = cbq + HIDn;
  float* ctx = cbk + HIDn;                  // [M,HID]
  float* t1  = ctx + (size_t)M * HIDn;      // pre-LN1
  float* h   = t1  + (size_t)M * HIDn;
  float* u   = h   + (size_t)M * HIDn;      // [M,FF]
  float* t2  = u   + (size_t)M * FFn;       // pre-LN2

  const dim3 blk(256);
  const dim3 gHH(HIDn / NT, HIDn / MT);     // 256x256 GEMMs
  const dim3 gMH(HIDn / NT, M / MT);        // 1024x256 GEMMs
  const dim3 gMF(FFn / NT, M / MT);         // 1024x1024 GEMM

  // Composed attack weights: aq = x@(Wq@AWq) + (bq@AWq + Abq)
  gemm_wmma_kernel<<<gHH, blk, 0, stream>>>(Wq, HIDn, AWq, HIDn, nullptr, nullptr, 0,
                                            CWq, HIDn, HIDn, 0);
  gemm_wmma_kernel<<<gHH, blk, 0, stream>>>(Wk, HIDn, AWk, HIDn, nullptr, nullptr, 0,
                                            CWk, HIDn, HIDn, 0);
  bias_compose_kernel<<<1, HIDn, 0, stream>>>(bq, AWq, Abq, cbq);
  bias_compose_kernel<<<1, HIDn, 0, stream>>>(bk, AWk, Abk, cbk);

  // Five projections straight from x
  gemm_wmma_kernel<<<gMH, blk, 0, stream>>>(x, HIDn, Wq,  HIDn, bq,  nullptr, 0, q,  HIDn, HIDn, 0);
  gemm_wmma_kernel<<<gMH, blk, 0, stream>>>(x, HIDn, Wk,  HIDn, bk,  nullptr, 0, k,  HIDn, HIDn, 0);
  gemm_wmma_kernel<<<gMH, blk, 0, stream>>>(x, HIDn, Wv,  HIDn, bv,  nullptr, 0, v,  HIDn, HIDn, 0);
  gemm_wmma_kernel<<<gMH, blk, 0, stream>>>(x, HIDn, CWq, HIDn, cbq, nullptr, 0, aq, HIDn, HIDn, 0);
  gemm_wmma_kernel<<<gMH, blk, 0, stream>>>(x, HIDn, CWk, HIDn, cbk, nullptr, 0, ak, HIDn, HIDn, 0);

  // Fused attention (32 workgroups, 198 KB LDS each — CDNA5 allows 320 KB/WGP)
  const size_t smem = (size_t)(3 * Sn * Sn + 4 * Sn) * sizeof(float);
  attention_kernel<<<Bn * Hn, 256, smem, stream>>>(q, k, v, aq, ak, mask, order_w,
                                                   order_b, dist_w, dist_b, scalar, ctx);

  // Output projection + residual, LN1
  gemm_wmma_kernel<<<gMH, blk, 0, stream>>>(ctx, HIDn, Wd, HIDn, bd, x, HIDn, t1, HIDn, HIDn, 0);
  layernorm_kernel<<<M, HIDn, 0, stream>>>(t1, g1, be1, h);

  // FFN: gelu(h@W1+b1)@W2+b2 + h, LN2 -> out
  gemm_wmma_kernel<<<gMF, blk, 0, stream>>>(h, HIDn, W1, FFn, b1, nullptr, 0, u, FFn, HIDn, 1);
  gemm_wmma_kernel<<<gMH, blk, 0, stream>>>(u, FFn, W2, HIDn, b2, h, HIDn, t2, HIDn, FFn, 0);
  layernorm_kernel<<<M, HIDn, 0, stream>>>(t2, g2, be2, out);

  (void)in_sizes; (void)n_in; (void)out_size; (void)ws_size;
}